// MatchingLayer_30425548324922
// MI455X (gfx1250) — compile-verified
//
#include <hip/hip_runtime.h>
#include <math.h>

#define B_  32
#define S_  64
#define H_  200
#define L_  20
#define CH_ (2*H_)           // 400, row stride of con_p/con_h
#define EPSF 1e-8f
#define OUTW 160             // 8*L per token
#define MV_H_OFF ((size_t)B_*S_*OUTW)
#define NEG_BIG (-3.402823466e38f)

typedef __attribute__((ext_vector_type(2))) float v2f;
typedef __attribute__((ext_vector_type(8))) float v8f;

// D = A(16x4,f32) * B(4x16,f32) + C(16x16,f32)  -- CDNA5 fp32 WMMA
static __device__ inline v8f wmma4(v2f a, v2f b, v8f c) {
  return __builtin_amdgcn_wmma_f32_16x16x4_f32(
      /*neg_a=*/false, a, /*neg_b=*/false, b,
      /*c_mod=*/(short)0, c, /*reuse_a=*/false, /*reuse_b=*/false);
}

// CDNA5 async global->LDS (ASYNCcnt). LDS offset = low 32 bits of the
// generic shared-aperture address (ISA 10.2 aperture mapping).
static __device__ inline void async_ld_b64(void* lds, const void* gl) {
  asm volatile("global_load_async_to_lds_b64 %0, %1, off"
               :: "v"((uint32_t)(uintptr_t)lds), "v"((uint64_t)(uintptr_t)gl)
               : "memory");
}
static __device__ inline void async_ld_b32(void* lds, const void* gl) {
  asm volatile("global_load_async_to_lds_b32 %0, %1, off"
               :: "v"((uint32_t)(uintptr_t)lds), "v"((uint64_t)(uintptr_t)gl)
               : "memory");
}
#define S_WAIT_ASYNC(N) asm volatile("s_wait_asynccnt " #N ::: "memory")

#define KCH   20               // K-chunk (floats) for 64x64 Gram kernels
#define NCH   (H_ / KCH)       // 10 chunks
#define APAD  26               // LDS row pitch (104B: 8B aligned, conflict-free)

// ---------------------------------------------------------------------------
// Kernel 1: cosine attention matrices att[dir][b][p][q] + row/col sums.
// Double-buffered async global->LDS staging feeding the fp32 WMMA loop.
// grid = 64 (b*2+dir), block = 256 (8 waves)
// ---------------------------------------------------------------------------
__global__ __launch_bounds__(256) void attention_kernel(
    const float* __restrict__ con_p, const float* __restrict__ con_h,
    float* __restrict__ att, float* __restrict__ rowsum, float* __restrict__ colsum)
{
  __shared__ float sstage[2][2][S_][APAD];   // [buf][0=p,1=h][row][k-chunk]
  __shared__ float sdot[S_][S_ + 1];
  __shared__ float snp[S_], snh[S_];
  const int blk = blockIdx.x;
  const int b = blk >> 1, dir = blk & 1;
  const float* p = con_p + (size_t)b * S_ * CH_ + dir * H_;
  const float* h = con_h + (size_t)b * S_ * CH_ + dir * H_;
  const int tid = threadIdx.x;
  const int wave = tid >> 5, lane = tid & 31;
  const int half = lane >> 4, r = lane & 15;

  // 5 async b64 loads per thread per chunk (64 rows x 10 pairs x 2 matrices)
  auto issue = [&](int c, int buf) {
#pragma unroll
    for (int j = 0; j < 5; ++j) {
      const int i = tid + j * 256;          // 0..1279
      const int mat = (i >= 640);
      const int idx = mat ? i - 640 : i;
      const int row = idx / 10, g = idx % 10;
      const float* src = (mat ? h : p) + (size_t)row * CH_ + c * KCH + g * 2;
      async_ld_b64(&sstage[buf][mat][row][g * 2], src);
    }
  };

  issue(0, 0);

  // norms overlap the first chunk's async loads
  if (tid < 64) {
    const float* row = p + (size_t)tid * CH_;
    float s = 0.f;
    for (int k = 0; k < H_; ++k) { const float v = row[k]; s += v * v; }
    snp[tid] = sqrtf(s);
  } else if (tid < 128) {
    const float* row = h + (size_t)(tid - 64) * CH_;
    float s = 0.f;
    for (int k = 0; k < H_; ++k) { const float v = row[k]; s += v * v; }
    snh[tid - 64] = sqrtf(s);
  }

  v8f acc0 = {0.f,0.f,0.f,0.f,0.f,0.f,0.f,0.f};
  v8f acc1 = acc0;
  const int t0 = wave, t1 = wave + 8;
  const int tm0 = t0 >> 2, tn0 = t0 & 3, tm1 = t1 >> 2, tn1 = t1 & 3;

  for (int c = 0; c < NCH; ++c) {
    if (c + 1 < NCH) { issue(c + 1, (c + 1) & 1); S_WAIT_ASYNC(5); }
    else             { S_WAIT_ASYNC(0); }
    __syncthreads();
    const float (*A)[APAD] = sstage[c & 1][0];
    const float (*Bm)[APAD] = sstage[c & 1][1];
#pragma unroll
    for (int kk = 0; kk < KCH / 4; ++kk) {
      const int k = kk * 4 + 2 * half;
      v2f a0 = *(const v2f*)&A[tm0 * 16 + r][k];
      v2f b0 = *(const v2f*)&Bm[tn0 * 16 + r][k];
      acc0 = wmma4(a0, b0, acc0);
      v2f a1 = *(const v2f*)&A[tm1 * 16 + r][k];
      v2f b1 = *(const v2f*)&Bm[tn1 * 16 + r][k];
      acc1 = wmma4(a1, b1, acc1);
    }
    __syncthreads();
  }

#pragma unroll
  for (int i = 0; i < 8; ++i) {
    sdot[tm0 * 16 + i + 8 * half][tn0 * 16 + r] = acc0[i];
    sdot[tm1 * 16 + i + 8 * half][tn1 * 16 + r] = acc1[i];
  }
  __syncthreads();

  for (int i = tid; i < S_ * S_; i += 256) {
    const int pp = i >> 6, qq = i & 63;
    const float d = snp[pp] * snh[qq];
    const float den = d > EPSF ? d : EPSF;
    sdot[pp][qq] = sdot[pp][qq] / den;
  }
  __syncthreads();

  float* attb = att + ((size_t)dir * B_ + b) * S_ * S_;
  if (tid < 64) {
    float s = 0.f;
    for (int q = 0; q < S_; ++q) { const float v = sdot[tid][q]; s += v; attb[(size_t)tid * S_ + q] = v; }
    rowsum[((size_t)dir * B_ + b) * S_ + tid] = s;
  } else if (tid < 128) {
    const int q = tid - 64;
    float s = 0.f;
    for (int pp = 0; pp < S_; ++pp) s += sdot[pp][q];
    colsum[((size_t)dir * B_ + b) * S_ + q] = s;
  }
}

// ---------------------------------------------------------------------------
// Kernel 2: per N-tile: stage h/p panels (async) -> WMMA att@h and att^T@p
// from LDS, plus att-max products from the same staged panels.
// grid = 64 (b*2+dir), block = 256
// ---------------------------------------------------------------------------
#define NPAD 18
__global__ __launch_bounds__(256) void att_products_kernel(
    const float* __restrict__ con_p, const float* __restrict__ con_h,
    const float* __restrict__ att, const float* __restrict__ rowsum,
    const float* __restrict__ colsum,
    float* __restrict__ mean_h, float* __restrict__ mean_p,
    float* __restrict__ max_h, float* __restrict__ max_p)
{
  __shared__ float satt[S_][S_ + 1];
  __shared__ float pan[2][S_][NPAD];     // [0]=h-panel, [1]=p-panel (16 cols)
  __shared__ float srs[S_], scs[S_];
  const int blk = blockIdx.x;
  const int b = blk >> 1, dir = blk & 1;
  const float* p = con_p + (size_t)b * S_ * CH_ + dir * H_;
  const float* h = con_h + (size_t)b * S_ * CH_ + dir * H_;
  const float* attb = att + ((size_t)dir * B_ + b) * S_ * S_;
  const int tid = threadIdx.x;

  // async-stage the 64x64 att matrix into padded LDS
#pragma unroll
  for (int j = 0; j < 16; ++j) {
    const int i = tid + j * 256;           // 0..4095
    async_ld_b32(&satt[i >> 6][i & 63], attb + i);
  }
  if (tid < 64) srs[tid] = rowsum[((size_t)dir * B_ + b) * S_ + tid];
  else if (tid < 128) scs[tid - 64] = colsum[((size_t)dir * B_ + b) * S_ + tid - 64];

  const int wave = tid >> 5, lane = tid & 31;
  const int half = lane >> 4, r = lane & 15;
  const int isP = wave >= 4;               // waves 0-3: att@h, 4-7: att^T@p
  const int tm = wave & 3;
  const size_t base = ((size_t)dir * B_ + b) * S_ * H_;
  const int NT = (H_ + 15) / 16;           // 13

  for (int tn = 0; tn < NT; ++tn) {
    // stage both panels: 4 b64 per thread (2 matrices x 64 rows x 8 pairs)
#pragma unroll
    for (int j = 0; j < 4; ++j) {
      const int i = tid + j * 256;         // 0..1023
      const int mat = i >> 9;              // 0=h, 1=p
      const int idx = i & 511;
      const int row = idx >> 3, g = idx & 7;
      const int n = tn * 16 + g * 2;
      if (n < H_) {
        async_ld_b64(&pan[mat][row][g * 2], (mat ? p : h) + (size_t)row * CH_ + n);
      } else {
        pan[mat][row][g * 2] = 0.f; pan[mat][row][g * 2 + 1] = 0.f;
      }
    }
    S_WAIT_ASYNC(0);
    __syncthreads();

    v8f c = {0.f,0.f,0.f,0.f,0.f,0.f,0.f,0.f};
#pragma unroll 4
    for (int k0 = 0; k0 < S_; k0 += 4) {
      const int k = k0 + 2 * half;
      v2f a, bb;
      if (!isP) { a.x = satt[tm * 16 + r][k]; a.y = satt[tm * 16 + r][k + 1]; }
      else      { a.x = satt[k][tm * 16 + r]; a.y = satt[k + 1][tm * 16 + r]; }
      bb = *(const v2f*)&pan[isP][k][r];   // pan[k][r], pan[k+1][r] pitch NPAD
      bb.y = pan[isP][k + 1][r];
      bb.x = pan[isP][k][r];
      c = wmma4(a, bb, c);
    }
#pragma unroll
    for (int i = 0; i < 8; ++i) {
      const int row = tm * 16 + i + 8 * half;
      const int col = tn * 16 + r;
      if (col < H_) {
        const float s = isP ? scs[row] : srs[row];
        const float den = s > EPSF ? s : EPSF;
        (isP ? mean_p : mean_h)[base + (size_t)row * H_ + col] = c[i] / den;
      }
    }

    // att-max products for this N-tile, from the staged LDS panels
    for (int i = tid; i < S_ * 16; i += 256) {
      const int pp = i >> 4, nl = i & 15;
      const int n = tn * 16 + nl;
      if (n < H_) {
        float m1 = NEG_BIG, m2 = NEG_BIG;
        for (int q = 0; q < S_; ++q) {
          m1 = fmaxf(m1, satt[pp][q] * pan[0][q][nl]);
          m2 = fmaxf(m2, satt[q][pp] * pan[1][q][nl]);
        }
        max_h[base + (size_t)pp * H_ + n] = m1;
        max_p[base + (size_t)pp * H_ + n] = m2;
      }
    }
    __syncthreads();   // protect pan before next tile's staging
  }
}

// ---------------------------------------------------------------------------
// Kernel 3: per (b,l,dir) weighted 64x64 cosine Gram (WMMA, async-staged,
// double-buffered), then row/col maxes -> "max" slots.
// grid = 32*20*2 = 1280, block = 256
// ---------------------------------------------------------------------------
__global__ __launch_bounds__(256) void pairwise_kernel(
    const float* __restrict__ con_p, const float* __restrict__ con_h,
    const float* __restrict__ w3, const float* __restrict__ w4,
    float* __restrict__ out)
{
  __shared__ float sstage[2][2][S_][APAD];
  __shared__ float sdot[S_][S_ + 1];
  __shared__ float sw[H_];
  __shared__ float snp[S_], snh[S_];
  const int blk = blockIdx.x;
  const int dir = blk & 1;
  const int l = (blk >> 1) % L_;
  const int b = blk / (2 * L_);
  const float* w = (dir == 0 ? w3 : w4) + (size_t)l * H_;
  const float* p = con_p + (size_t)b * S_ * CH_ + dir * H_;
  const float* h = con_h + (size_t)b * S_ * CH_ + dir * H_;
  const int tid = threadIdx.x;
  const int wave = tid >> 5, lane = tid & 31;
  const int half = lane >> 4, r = lane & 15;

  auto issue = [&](int c, int buf) {
#pragma unroll
    for (int j = 0; j < 5; ++j) {
      const int i = tid + j * 256;
      const int mat = (i >= 640);
      const int idx = mat ? i - 640 : i;
      const int row = idx / 10, g = idx % 10;
      const float* src = (mat ? h : p) + (size_t)row * CH_ + c * KCH + g * 2;
      async_ld_b64(&sstage[buf][mat][row][g * 2], src);
    }
  };

  issue(0, 0);
  for (int i = tid; i < H_; i += 256) sw[i] = w[i];

  if (tid < 64) {
    const float* row = p + (size_t)tid * CH_;
    float s = 0.f;
    for (int k = 0; k < H_; ++k) { const float v = row[k] * w[k]; s += v * v; }
    snp[tid] = sqrtf(s);
  } else if (tid < 128) {
    const float* row = h + (size_t)(tid - 64) * CH_;
    float s = 0.f;
    for (int k = 0; k < H_; ++k) { const float v = row[k] * w[k]; s += v * v; }
    snh[tid - 64] = sqrtf(s);
  }

  v8f acc0 = {0.f,0.f,0.f,0.f,0.f,0.f,0.f,0.f};
  v8f acc1 = acc0;
  const int t0 = wave, t1 = wave + 8;
  const int tm0 = t0 >> 2, tn0 = t0 & 3, tm1 = t1 >> 2, tn1 = t1 & 3;

  for (int c = 0; c < NCH; ++c) {
    if (c + 1 < NCH) { issue(c + 1, (c + 1) & 1); S_WAIT_ASYNC(5); }
    else             { S_WAIT_ASYNC(0); }
    __syncthreads();
    const float (*A)[APAD] = sstage[c & 1][0];
    const float (*Bm)[APAD] = sstage[c & 1][1];
#pragma unroll
    for (int kk = 0; kk < KCH / 4; ++kk) {
      const int k = kk * 4 + 2 * half;
      const float w0 = sw[c * KCH + k], w1 = sw[c * KCH + k + 1];
      v2f a0, b0, a1, b1;
      a0.x = A[tm0 * 16 + r][k] * w0;  a0.y = A[tm0 * 16 + r][k + 1] * w1;
      b0.x = Bm[tn0 * 16 + r][k] * w0; b0.y = Bm[tn0 * 16 + r][k + 1] * w1;
      acc0 = wmma4(a0, b0, acc0);
      a1.x = A[tm1 * 16 + r][k] * w0;  a1.y = A[tm1 * 16 + r][k + 1] * w1;
      b1.x = Bm[tn1 * 16 + r][k] * w0; b1.y = Bm[tn1 * 16 + r][k + 1] * w1;
      acc1 = wmma4(a1, b1, acc1);
    }
    __syncthreads();
  }

#pragma unroll
  for (int i = 0; i < 8; ++i) {
    sdot[tm0 * 16 + i + 8 * half][tn0 * 16 + r] = acc0[i];
    sdot[tm1 * 16 + i + 8 * half][tn1 * 16 + r] = acc1[i];
  }
  __syncthreads();

  const int slot = (dir == 0 ? 20 : 100) + l;
  if (tid < 64) {
    const int pp = tid;
    float m = NEG_BIG;
    for (int q = 0; q < S_; ++q) {
      const float d = snp[pp] * snh[q];
      const float den = d > EPSF ? d : EPSF;
      m = fmaxf(m, sdot[pp][q] / den);
    }
    out[((size_t)b * S_ + pp) * OUTW + slot] = m;
  } else if (tid < 128) {
    const int q = tid - 64;
    float m = NEG_BIG;
    for (int pp = 0; pp < S_; ++pp) {
      const float d = snp[pp] * snh[q];
      const float den = d > EPSF ? d : EPSF;
      m = fmaxf(m, sdot[pp][q] / den);
    }
    out[MV_H_OFF + ((size_t)b * S_ + q) * OUTW + slot] = m;
  }
}

// ---------------------------------------------------------------------------
// Kernel 4: the 12 _mp_match cosine reductions. One thread per (var,b,s,l).
// ---------------------------------------------------------------------------
__global__ __launch_bounds__(256) void mp_match_kernel(
    const float* __restrict__ con_p, const float* __restrict__ con_h,
    const float* __restrict__ w1, const float* __restrict__ w2,
    const float* __restrict__ w5, const float* __restrict__ w6,
    const float* __restrict__ w7, const float* __restrict__ w8,
    const float* __restrict__ mean_h, const float* __restrict__ mean_p,
    const float* __restrict__ max_h, const float* __restrict__ max_p,
    float* __restrict__ out)
{
  const int idx = blockIdx.x * blockDim.x + threadIdx.x;
  const int total = 12 * B_ * S_ * L_;
  if (idx >= total) return;
  const int l = idx % L_;
  const int s = (idx / L_) % S_;
  const int b = (idx / (L_ * S_)) % B_;
  const int var = idx / (L_ * S_ * B_);

  auto prow = [&](int ss, int d) { return con_p + ((size_t)b * S_ + ss) * CH_ + d * H_; };
  auto hrow = [&](int ss, int d) { return con_h + ((size_t)b * S_ + ss) * CH_ + d * H_; };
  auto wrow = [&](const float* arr, int d) { return arr + (((size_t)d * B_ + b) * S_ + s) * H_; };

  const float* v1; const float* v2; const float* w; int side; int slot;
  switch (var) {
    case 0:  v1 = prow(s,0); v2 = hrow(S_-1,0);      w = w1; side = 0; slot = 0;   break;
    case 1:  v1 = prow(s,1); v2 = hrow(0,1);         w = w2; side = 0; slot = 80;  break;
    case 2:  v1 = hrow(s,0); v2 = prow(S_-1,0);      w = w1; side = 1; slot = 0;   break;
    case 3:  v1 = hrow(s,1); v2 = prow(0,1);         w = w2; side = 1; slot = 80;  break;
    case 4:  v1 = prow(s,0); v2 = wrow(mean_h,0);    w = w5; side = 0; slot = 40;  break;
    case 5:  v1 = prow(s,1); v2 = wrow(mean_h,1);    w = w6; side = 0; slot = 120; break;
    case 6:  v1 = hrow(s,0); v2 = wrow(mean_p,0);    w = w5; side = 1; slot = 40;  break;
    case 7:  v1 = hrow(s,1); v2 = wrow(mean_p,1);    w = w6; side = 1; slot = 120; break;
    case 8:  v1 = prow(s,0); v2 = wrow(max_h,0);     w = w7; side = 0; slot = 60;  break;
    case 9:  v1 = prow(s,1); v2 = wrow(max_h,1);     w = w8; side = 0; slot = 140; break;
    case 10: v1 = hrow(s,0); v2 = wrow(max_p,0);     w = w7; side = 1; slot = 60;  break;
    default: v1 = hrow(s,1); v2 = wrow(max_p,1);     w = w8; side = 1; slot = 140; break;
  }

  const float* wl = w + (size_t)l * H_;
  float dot = 0.f, na2 = 0.f, nb2 = 0.f;
  for (int k = 0; k < H_; ++k) {
    const float wv = wl[k];
    const float a = v1[k] * wv;
    const float bb = v2[k] * wv;
    dot += a * bb; na2 += a * a; nb2 += bb * bb;
  }
  const float den = fmaxf(sqrtf(na2) * sqrtf(nb2), EPSF);
  const size_t o = (side == 0 ? (size_t)0 : MV_H_OFF) +
                   ((size_t)b * S_ + s) * OUTW + slot + l;
  out[o] = dot / den;
}

// ---------------------------------------------------------------------------
extern "C" void kernel_launch(void* const* d_in, const int* in_sizes, int n_in,
                              void* d_out, int out_size, void* d_ws, size_t ws_size,
                              hipStream_t stream) {
  (void)in_sizes; (void)n_in; (void)out_size; (void)ws_size;
  const float* con_p = (const float*)d_in[0];
  const float* con_h = (const float*)d_in[1];
  const float* w1 = (const float*)d_in[2];
  const float* w2 = (const float*)d_in[3];
  const float* w3 = (const float*)d_in[4];
  const float* w4 = (const float*)d_in[5];
  const float* w5 = (const float*)d_in[6];
  const float* w6 = (const float*)d_in[7];
  const float* w7 = (const float*)d_in[8];
  const float* w8 = (const float*)d_in[9];
  float* out = (float*)d_out;
  float* ws = (float*)d_ws;

  const size_t N_ATT = (size_t)2 * B_ * S_ * S_;   // 262144
  const size_t N_SUM = (size_t)2 * B_ * S_;        // 4096
  const size_t N_BSH = (size_t)2 * B_ * S_ * H_;   // 819200
  float* att    = ws;
  float* rowsum = att + N_ATT;
  float* colsum = rowsum + N_SUM;
  float* mean_h = colsum + N_SUM;
  float* mean_p = mean_h + N_BSH;
  float* max_h  = mean_p + N_BSH;
  float* max_p  = max_h + N_BSH;

  attention_kernel<<<2 * B_, 256, 0, stream>>>(con_p, con_h, att, rowsum, colsum);
  att_products_kernel<<<2 * B_, 256, 0, stream>>>(con_p, con_h, att, rowsum, colsum,
                                                  mean_h, mean_p, max_h, max_p);
  pairwise_kernel<<<2 * B_ * L_, 256, 0, stream>>>(con_p, con_h, w3, w4, out);
  const int total = 12 * B_ * S_ * L_;
  mp_match_kernel<<<(total + 255) / 256, 256, 0, stream>>>(
      con_p, con_h, w1, w2, w5, w6, w7, w8, mean_h, mean_p, max_h, max_p, out);
}